// world_state_encoder_18665927868454
// MI455X (gfx1250) — compile-verified
//
#include <hip/hip_runtime.h>
#include <stdint.h>

// World-state encoder gather for MI455X (gfx1250).
// Pure store-bandwidth problem: 448 MB f32 output at 23.3 TB/s ~= 19 us floor.
// Strategy: async-copy the 3.5 KB embedding table into LDS (CDNA5
// global_load_async_to_lds path), scalarize the per-row index load (wave32 ->
// SMEM), then stream ds_load_b128 -> non-temporal global_store_b128.

typedef float v4f __attribute__((ext_vector_type(4)));

constexpr int kBatch        = 32768;
constexpr int kRowsPerBatch = 28;            // 7 beakers * 4 color slots
constexpr int kDim          = 128;
constexpr int kRows         = kBatch * kRowsPerBatch;   // 917504 output rows
constexpr int kChunks       = kRows * (kDim / 4);       // float4 chunks: 29,360,128
constexpr int kTableFloats  = 7 * kDim;                 // 896 floats = 3.5 KB
constexpr int kBlockSize    = 256;                      // 8 wave32 per block
constexpr int kGridSize     = 2048;                     // 56 chunks per thread

__global__ __launch_bounds__(kBlockSize)
void ws_encoder_gather(const int* __restrict__ X,
                       const float* __restrict__ emb,
                       float* __restrict__ out) {
    __shared__ float table[kTableFloats];

    const int tid = threadIdx.x;

    // ---- Stage embedding table into LDS with CDNA5 async copy ----
    // 224 lanes (waves 0..6) each move 16 bytes; wave 7 skips uniformly.
    if (tid < kTableFloats / 4) {
        // Low 32 bits of a flat shared-memory address are the logical LDS
        // byte offset (ISA aperture mapping).
        unsigned lds_off = (unsigned)(uintptr_t)(&table[0]) + (unsigned)tid * 16u;
        uint64_t gaddr   = (uint64_t)(uintptr_t)emb + (uint64_t)tid * 16u;
        asm volatile("global_load_async_to_lds_b128 %0, %1, off"
                     :: "v"(lds_off), "v"(gaddr)
                     : "memory");
    }
#if __has_builtin(__builtin_amdgcn_s_wait_asynccnt)
    __builtin_amdgcn_s_wait_asynccnt(0);
#else
    asm volatile("s_wait_asynccnt 0" ::: "memory");
#endif
    __syncthreads();

    const int stride = (int)(gridDim.x * blockDim.x);
    int c = (int)(blockIdx.x * blockDim.x) + tid;

    // Main loop, unrolled x4 for MLP on the id -> LDS -> store chain.
    for (; c + 3 * stride < kChunks; c += 4 * stride) {
        int id[4];
#pragma unroll
        for (int u = 0; u < 4; ++u) {
            int cc  = c + u * stride;
            int row = cc >> 5;                       // 32 float4 chunks per row
            unsigned b = (unsigned)row / 28u;        // batch index (mul-hi)
            int s    = row - (int)b * 28;            // slot within batch row
            int xoff = (int)b * 35 + (s >> 2) * 5 + (s & 3) + 1;
            // Every lane of a wave shares one row -> force the scalar path.
            xoff  = __builtin_amdgcn_readfirstlane(xoff);
            id[u] = X[xoff];                         // 0..6
        }
        v4f v[4];
#pragma unroll
        for (int u = 0; u < 4; ++u) {
            int cc = c + u * stride;
            int q  = cc & 31;                        // float4 column in row
            v[u] = *(const v4f*)(&table[id[u] * kDim + q * 4]);  // ds_load_b128
        }
#pragma unroll
        for (int u = 0; u < 4; ++u) {
            int cc = c + u * stride;
            // NT store: don't rinse 448 MB through the 192 MB L2.
            __builtin_nontemporal_store(v[u], (v4f*)(out + (size_t)cc * 4));
        }
    }
    // Tail (empty for the default 2048x256 launch, kept for safety).
    for (; c < kChunks; c += stride) {
        int row = c >> 5;
        unsigned b = (unsigned)row / 28u;
        int s    = row - (int)b * 28;
        int xoff = (int)b * 35 + (s >> 2) * 5 + (s & 3) + 1;
        xoff = __builtin_amdgcn_readfirstlane(xoff);
        int id = X[xoff];
        int q  = c & 31;
        v4f v = *(const v4f*)(&table[id * kDim + q * 4]);
        __builtin_nontemporal_store(v, (v4f*)(out + (size_t)c * 4));
    }
}

extern "C" void kernel_launch(void* const* d_in, const int* in_sizes, int n_in,
                              void* d_out, int out_size, void* d_ws, size_t ws_size,
                              hipStream_t stream) {
    (void)in_sizes; (void)n_in; (void)out_size; (void)d_ws; (void)ws_size;
    const int*   X   = (const int*)d_in[0];     // [32768, 35] ids in 0..6
    const float* emb = (const float*)d_in[1];   // [7, 128] f32
    float*       out = (float*)d_out;           // [32768, 3584] f32
    ws_encoder_gather<<<kGridSize, kBlockSize, 0, stream>>>(X, emb, out);
}